// LocalCrossAttentionModule_37615323578698
// MI455X (gfx1250) — compile-verified
//
#include <hip/hip_runtime.h>
#include <hip/hip_bf16.h>

typedef __attribute__((ext_vector_type(2))) float v2f;
typedef __attribute__((ext_vector_type(8))) float v8f;

// Problem constants
#define BATCH   4
#define CKV     128
#define HH      432
#define WW      432
#define PP      5          // patch size
#define STRIDE  36
#define PI      12         // patches per dim
#define NPATCH  (PI*PI)    // 144
#define MROWS   (BATCH*NPATCH)   // 576
#define KDIM    (CKV*PP*PP)      // 3200
#define DFF     1600
#define OUTC    64
#define OCONV   128

// ---------------------------------------------------------------------------
// Kernel 1: fill y with conv_b broadcast (dominant 382MB write, float4 stores)
// ---------------------------------------------------------------------------
__global__ void fill_bias_kernel(const float* __restrict__ conv_b,
                                 float4* __restrict__ y4) {
    const int planeQuads = (HH * WW) / 4;                 // 46656 float4 per (b,o) plane
    long long idx = (long long)blockIdx.x * blockDim.x + threadIdx.x;
    const long long total = (long long)BATCH * OCONV * planeQuads;
    if (idx >= total) return;
    int o = (int)((idx / planeQuads) % OCONV);
    float v = conv_b[o];
    y4[idx] = make_float4(v, v, v, v);
}

// ---------------------------------------------------------------------------
// Kernel 2: gather kvf patches -> dense KVF[576][3200], zero pad at edges
// ---------------------------------------------------------------------------
__global__ void gather_kvf_kernel(const float* __restrict__ x_kv,
                                  float* __restrict__ KVF) {
    long long idx = (long long)blockIdx.x * blockDim.x + threadIdx.x;
    const long long total = (long long)MROWS * KDIM;      // 1,843,200
    if (idx >= total) return;
    int row = (int)(idx / KDIM);
    int k   = (int)(idx % KDIM);
    int c   = k / (PP * PP);
    int rem = k % (PP * PP);
    int p   = rem / PP;
    int q   = rem % PP;
    int b   = row / NPATCH;
    int n   = row % NPATCH;
    int pi  = n / PI;
    int pj  = n % PI;
    int hr  = pi * STRIDE + p - 2;     // padded coord minus E
    int wc  = pj * STRIDE + q - 2;
    float val = 0.0f;
    if (hr >= 0 && hr < HH && wc >= 0 && wc < WW) {
        val = x_kv[(((long long)b * CKV + c) * HH + hr) * WW + wc];
    }
    KVF[idx] = val;
}

// ---------------------------------------------------------------------------
// Kernel 3: V = KVF @ Wv^T + bv   (576 x 1600, K=3200), fp32 WMMA 16x16x4.
// One wave (32 threads) per 16x64 output tile; 4 accumulators amortize A.
// A-lane layout (ISA 16x4 f32): lane l<16 holds A[m0+l][k0..k0+1],
// lane l>=16 holds A[m0+l-16][k0+2..k0+3] -> one float2 load per lane.
// B mirrors with lane selecting column n, half selecting K pair.
// C/D layout: vgpr r, lane l -> row m0 + r + 8*(l>>4), col n0 + (l&15).
// ---------------------------------------------------------------------------
__global__ void v_gemm_wmma_kernel(const float* __restrict__ KVF,
                                   const float* __restrict__ Wv,
                                   const float* __restrict__ bv,
                                   float* __restrict__ V) {
    const int lane = threadIdx.x;          // blockDim.x == 32, EXEC all ones
    const int half = lane >> 4;            // 0 or 1
    const int lrow = lane & 15;

    const int m0 = blockIdx.y * 16;        // 36 tiles
    const int n0 = blockIdx.x * 64;        // 25 tiles

    const float* aPtr = KVF + (long long)(m0 + lrow) * KDIM + 2 * half;
    const float* bPtr0 = Wv + (long long)(n0 +  0 + lrow) * KDIM + 2 * half;
    const float* bPtr1 = Wv + (long long)(n0 + 16 + lrow) * KDIM + 2 * half;
    const float* bPtr2 = Wv + (long long)(n0 + 32 + lrow) * KDIM + 2 * half;
    const float* bPtr3 = Wv + (long long)(n0 + 48 + lrow) * KDIM + 2 * half;

    v8f acc0 = {}, acc1 = {}, acc2 = {}, acc3 = {};

    for (int k0 = 0; k0 < KDIM; k0 += 4) {
        v2f a  = *(const v2f*)(aPtr  + k0);
        v2f b0 = *(const v2f*)(bPtr0 + k0);
        v2f b1 = *(const v2f*)(bPtr1 + k0);
        v2f b2 = *(const v2f*)(bPtr2 + k0);
        v2f b3 = *(const v2f*)(bPtr3 + k0);
        acc0 = __builtin_amdgcn_wmma_f32_16x16x4_f32(false, a, false, b0,
                                                     (short)0, acc0, false, false);
        acc1 = __builtin_amdgcn_wmma_f32_16x16x4_f32(false, a, false, b1,
                                                     (short)0, acc1, false, false);
        acc2 = __builtin_amdgcn_wmma_f32_16x16x4_f32(false, a, false, b2,
                                                     (short)0, acc2, false, false);
        acc3 = __builtin_amdgcn_wmma_f32_16x16x4_f32(false, a, false, b3,
                                                     (short)0, acc3, false, false);
    }

    // Add bias (depends on column only) and store D tiles.
    float bias0 = bv[n0 +  0 + lrow];
    float bias1 = bv[n0 + 16 + lrow];
    float bias2 = bv[n0 + 32 + lrow];
    float bias3 = bv[n0 + 48 + lrow];
#pragma unroll
    for (int r = 0; r < 8; ++r) {
        int m = m0 + r + 8 * half;
        float* vrow = V + (long long)m * DFF + n0 + lrow;
        vrow[ 0] = acc0[r] + bias0;
        vrow[16] = acc1[r] + bias1;
        vrow[32] = acc2[r] + bias2;
        vrow[48] = acc3[r] + bias3;
    }
}

// ---------------------------------------------------------------------------
// Kernel 4: for each patch pixel, y[b,o,h,w] = conv_w[o,:] . v[row, :, p, q]
//           + conv_b[o]. One 256-thread block per pixel (14400 blocks total).
// ---------------------------------------------------------------------------
__global__ void scatter_conv_kernel(const float* __restrict__ V,
                                    const float* __restrict__ conv_w,
                                    const float* __restrict__ conv_b,
                                    float* __restrict__ y) {
    __shared__ float vsh[OUTC];
    int pix = blockIdx.x;                  // 0 .. 14399
    int row = pix / (PP * PP);             // (b, n)
    int pos = pix % (PP * PP);
    int p = pos / PP, q = pos % PP;
    int b = row / NPATCH;
    int n = row % NPATCH;
    int pi = n / PI, pj = n % PI;
    int h = pi * STRIDE + p;
    int w = pj * STRIDE + q;

    int tid = threadIdx.x;                 // 0 .. 255
    if (tid < OUTC) {
        vsh[tid] = V[(long long)row * DFF + tid * (PP * PP) + pos];
    }
    __syncthreads();

    float s = conv_b[tid];
    const float* wrow = conv_w + tid * OUTC;
#pragma unroll 8
    for (int c = 0; c < OUTC; ++c) {
        s = fmaf(wrow[c], vsh[c], s);
    }
    y[(((long long)b * OCONV + tid) * HH + h) * WW + w] = s;
}

// ---------------------------------------------------------------------------
// Launch
// ---------------------------------------------------------------------------
extern "C" void kernel_launch(void* const* d_in, const int* in_sizes, int n_in,
                              void* d_out, int out_size, void* d_ws, size_t ws_size,
                              hipStream_t stream) {
    // setup_inputs order:
    // 0 x_query, 1 x_kv, 2 Wq, 3 bq, 4 Wk, 5 bk, 6 Wv, 7 bv, 8 conv_w, 9 conv_b
    const float* x_kv   = (const float*)d_in[1];
    const float* Wv     = (const float*)d_in[6];
    const float* bv     = (const float*)d_in[7];
    const float* conv_w = (const float*)d_in[8];
    const float* conv_b = (const float*)d_in[9];
    float* y = (float*)d_out;

    // Workspace layout: KVF (576*3200 f32 = 7.37MB), V (576*1600 f32 = 3.69MB)
    float* KVF = (float*)d_ws;
    float* V   = KVF + (long long)MROWS * KDIM;

    // 1) dominant output fill: 23,887,872 float4 stores
    {
        long long total4 = (long long)BATCH * OCONV * (HH * WW / 4);
        int threads = 256;
        int blocks = (int)((total4 + threads - 1) / threads);
        fill_bias_kernel<<<blocks, threads, 0, stream>>>(conv_b, (float4*)y);
    }
    // 2) gather patches into dense KVF
    {
        long long total = (long long)MROWS * KDIM;
        int threads = 256;
        int blocks = (int)((total + threads - 1) / threads);
        gather_kvf_kernel<<<blocks, threads, 0, stream>>>(x_kv, KVF);
    }
    // 3) WMMA fp32 GEMM: V = KVF @ Wv^T + bv
    {
        dim3 grid(DFF / 64, MROWS / 16);   // (25, 36)
        v_gemm_wmma_kernel<<<grid, 32, 0, stream>>>(KVF, Wv, bv, V);
    }
    // 4) fused conv + scatter of the 14400 patch pixels
    {
        scatter_conv_kernel<<<MROWS * PP * PP, OCONV * 2, 0, stream>>>(
            V, conv_w, conv_b, y);
    }
}